// myTECA_88158498718009
// MI455X (gfx1250) — compile-verified
//
#include <hip/hip_runtime.h>
#include <math.h>

typedef __attribute__((ext_vector_type(2))) float v2f;
typedef __attribute__((ext_vector_type(8))) float v8f;

#define B_  32
#define C_  32
#define H_  256
#define W_  256
#define L_  512
#define HW_ (H_*W_)

// ---------------------------------------------------------------------------
// float atomic-max via int/uint bit trick (correct for all sign combinations)
// ---------------------------------------------------------------------------
__device__ __forceinline__ void atomicMaxF(float* addr, float val) {
    if (val >= 0.0f) atomicMax((int*)addr, __float_as_int(val));
    else             atomicMin((unsigned int*)addr, __float_as_uint(val));
}

// ---------------------------------------------------------------------------
// init workspace targets of atomics (colsum = 0, colmax = -inf)
// grid: 1024 blocks x 256 -> 262144 elements each
// ---------------------------------------------------------------------------
__global__ __launch_bounds__(256) void init_kernel(float* __restrict__ colsum,
                                                   float* __restrict__ colmax) {
    int i = blockIdx.x * 256 + threadIdx.x;
    colsum[i] = 0.0f;
    colmax[i] = -INFINITY;
}

// ---------------------------------------------------------------------------
// Pass 1: single streaming read of x producing
//   rowsum/rowmax [b,c,h]   (direct write)
//   colsum/colmax [b,c,w]   (atomic over 16 h-tiles)
//   pool [b, {max,mean}, h, w] (direct write)
// grid: B_*16 blocks (one per (b, 16-row tile)), 256 threads (thread = w)
// ---------------------------------------------------------------------------
__global__ __launch_bounds__(256) void reduce_kernel(
    const float* __restrict__ x,
    float* __restrict__ rowsum, float* __restrict__ rowmax,
    float* __restrict__ colsum, float* __restrict__ colmax,
    float* __restrict__ pool) {
    const int b   = blockIdx.x >> 4;
    const int h0  = (blockIdx.x & 15) * 16;
    const int tid = threadIdx.x;            // = w
    __shared__ float lds[16 * 256];

    float chs[16], chm[16];
#pragma unroll
    for (int r = 0; r < 16; ++r) { chs[r] = 0.0f; chm[r] = -INFINITY; }

    for (int c = 0; c < C_; ++c) {
        const float* xp = x + ((b * C_ + c) * H_ + h0) * W_ + tid;
        float cs = 0.0f, cm = -INFINITY;
#pragma unroll
        for (int r = 0; r < 16; ++r) {
            float t = xp[r * W_];
            cs += t;                cm = fmaxf(cm, t);
            chs[r] += t;            chm[r] = fmaxf(chm[r], t);
            lds[r * 256 + tid] = t;
        }
        // column (over-h) partials for this 16-row tile
        atomicAdd(&colsum[(b * C_ + c) * W_ + tid], cs);
        atomicMaxF(&colmax[(b * C_ + c) * W_ + tid], cm);
        __syncthreads();

        // row (over-w) reductions: 8 waves x 2 half-waves -> 16 rows
        {
            const int wid  = tid >> 5;
            const int lane = tid & 31;
            const int row  = wid * 2 + (lane >> 4);
            const int l16  = lane & 15;
            float s = 0.0f, mx = -INFINITY;
#pragma unroll
            for (int q = 0; q < 16; ++q) {
                float t = lds[row * 256 + l16 * 16 + q];
                s += t; mx = fmaxf(mx, t);
            }
#pragma unroll
            for (int off = 1; off < 16; off <<= 1) {
                s  += __shfl_xor(s, off, 32);
                mx  = fmaxf(mx, __shfl_xor(mx, off, 32));
            }
            if (l16 == 0) {
                rowsum[(b * C_ + c) * H_ + h0 + row] = s;
                rowmax[(b * C_ + c) * H_ + h0 + row] = mx;
            }
        }
        __syncthreads();
    }
    // channel stats -> pool[b][0]=max over c, pool[b][1]=mean over c
    float* pb = pool + (size_t)b * 2 * HW_;
#pragma unroll
    for (int r = 0; r < 16; ++r) {
        pb[(h0 + r) * W_ + tid]        = chm[r];
        pb[HW_ + (h0 + r) * W_ + tid]  = chs[r] * (1.0f / (float)C_);
    }
}

// ---------------------------------------------------------------------------
// Pass 2: conv01 as WMMA f32 16x16x4 GEMM.
// Per b: Y2[32,512] = Wm[32,64] x Y[64,512],
//   Wm[o][i*2+k] = w01[o,i,0,k];  Y[i*2+k][l] = y1[b,i,l,k]
// grid: B_*8 blocks x 256 threads (8 waves, 1 16x16 D-tile per wave)
// ---------------------------------------------------------------------------
__global__ __launch_bounds__(256) void conv01_wmma_kernel(
    const float* __restrict__ w01,
    const float* __restrict__ rowsum, const float* __restrict__ rowmax,
    const float* __restrict__ colsum, const float* __restrict__ colmax,
    float* __restrict__ y2) {
    const int b    = blockIdx.x >> 3;
    const int jb   = blockIdx.x & 7;
    const int wave = threadIdx.x >> 5;
    const int lane = threadIdx.x & 31;
    const int t    = jb * 8 + wave;     // 0..63 tiles
    const int mt   = t & 1;             // 2 M-tiles
    const int nt   = t >> 1;            // 32 N-tiles
    const int half = lane >> 4;
    const int l16  = lane & 15;
    const int n    = nt * 16 + l16;     // l in [0,512)

    // Y source for this lane's column (uniform per wave: n<256 or n>=256)
    const int   lw    = (n < H_) ? n : (n - H_);
    const float* meanP = (n < H_) ? rowsum : colsum;
    const float* maxP  = (n < H_) ? rowmax : colmax;

    v8f acc = {};
#pragma unroll
    for (int kk = 0; kk < 16; ++kk) {
        // A tile: Wm[mt*16+l16][kk*4 + 2*half + {0,1}]
        const int arow  = mt * 16 + l16;
        const int acol0 = kk * 4 + 2 * half;
        v2f a;
        a.x = w01[arow * 64 + acol0];
        a.y = w01[arow * 64 + acol0 + 1];
        // B tile: Y[kk*4 + 2*half + {0,1}][n]
        const int ik0 = kk * 4 + 2 * half;
        v2f bv;
        {
            int i0 = ik0 >> 1, k0 = ik0 & 1;
            int base0 = (b * C_ + i0) * 256 + lw;
            float v0 = k0 ? maxP[base0] : meanP[base0] * (1.0f / 256.0f);
            int ik1 = ik0 + 1;
            int i1 = ik1 >> 1, k1 = ik1 & 1;
            int base1 = (b * C_ + i1) * 256 + lw;
            float v1 = k1 ? maxP[base1] : meanP[base1] * (1.0f / 256.0f);
            bv.x = v0; bv.y = v1;
        }
        acc = __builtin_amdgcn_wmma_f32_16x16x4_f32(
            false, a, false, bv, (short)0, acc, false, false);
    }
    // D layout: VGPR j -> row (mt*16 + half*8 + j), col n
#pragma unroll
    for (int j = 0; j < 8; ++j) {
        int m = mt * 16 + half * 8 + j;
        y2[(b * C_ + m) * L_ + n] = acc[j];
    }
}

// ---------------------------------------------------------------------------
// Pass 3: depthwise conv1d(k=7,pad=3) + BN + h-swish + sigmoid -> a[b,c,512]
// grid: B_*C_ blocks x 256 threads (2 l-positions per thread)
// ---------------------------------------------------------------------------
__global__ __launch_bounds__(256) void post_kernel(
    const float* __restrict__ y2, const float* __restrict__ w02,
    const float* __restrict__ gamma, const float* __restrict__ beta,
    const float* __restrict__ bmean, const float* __restrict__ bvar,
    float* __restrict__ a) {
    const int bc = blockIdx.x;
    const int c  = bc & (C_ - 1);
    const int tid = threadIdx.x;
    __shared__ float row[L_ + 6];
    if (tid < 3)              row[tid] = 0.0f;
    if (tid >= 253 && tid < 256) row[L_ + 3 + (tid - 253)] = 0.0f;
    row[3 + tid]        = y2[bc * L_ + tid];
    row[3 + tid + 256]  = y2[bc * L_ + tid + 256];
    __syncthreads();

    const float scale = gamma[c] * rsqrtf(bvar[c] + 1e-5f);
    const float shift = beta[c] - bmean[c] * scale;
    float wts[7];
#pragma unroll
    for (int j = 0; j < 7; ++j) wts[j] = w02[c * 7 + j];

#pragma unroll
    for (int s = 0; s < 2; ++s) {
        const int l = tid + s * 256;
        float acc = 0.0f;
#pragma unroll
        for (int j = 0; j < 7; ++j) acc += row[l + j] * wts[j];
        float yn = acc * scale + shift;
        float hs = yn * fminf(fmaxf(yn + 3.0f, 0.0f), 6.0f) * (1.0f / 6.0f);
        a[bc * L_ + l] = 1.0f / (1.0f + __expf(-hs));
    }
}

// ---------------------------------------------------------------------------
// Pass 4: fused 7x7 spatial-attention conv + sigmoid + final product.
// grid: B_*(H_/4) blocks x 256 threads; thread = (row r of 4, 4 w's via float4)
// ---------------------------------------------------------------------------
__global__ __launch_bounds__(256) void fuse_out_kernel(
    const float* __restrict__ x, const float* __restrict__ a,
    const float* __restrict__ pool, const float* __restrict__ wsa,
    float* __restrict__ out) {
    const int b  = blockIdx.x >> 6;
    const int h0 = (blockIdx.x & 63) * 4;
    const int tid = threadIdx.x;
    const int r   = tid >> 6;       // 0..3
    const int wq  = tid & 63;       // float4 column

    __shared__ float ldsP[2 * 10 * 262];   // pool halo: rows h0-3..h0+6, cols -3..258
    __shared__ float ldsW[98];

    for (int idx = tid; idx < 2 * 10 * 262; idx += 256) {
        int ci  = idx / 2620;
        int rem = idx - ci * 2620;
        int rr  = rem / 262;
        int col = rem - rr * 262;
        int gh = h0 - 3 + rr;
        int gw = col - 3;
        float v = 0.0f;
        if (gh >= 0 && gh < H_ && gw >= 0 && gw < W_)
            v = pool[((size_t)b * 2 + ci) * HW_ + gh * W_ + gw];
        ldsP[idx] = v;
    }
    if (tid < 98) ldsW[tid] = wsa[tid];
    __syncthreads();

    // spatial attention for this thread's 4 pixels
    float sa4[4];
#pragma unroll
    for (int j = 0; j < 4; ++j) {
        const int w = wq * 4 + j;
        float acc = 0.0f;
#pragma unroll
        for (int ci = 0; ci < 2; ++ci)
#pragma unroll
            for (int dy = 0; dy < 7; ++dy)
#pragma unroll
                for (int dx = 0; dx < 7; ++dx)
                    acc += ldsP[ci * 2620 + (r + dy) * 262 + (w + dx)]
                         * ldsW[ci * 49 + dy * 7 + dx];
        sa4[j] = 1.0f / (1.0f + __expf(-acc));
    }

    const int h = h0 + r;
    for (int c = 0; c < C_; ++c) {
        const int rowbase = ((b * C_ + c) * H_ + h) * W_;
        const float* xrow = x + rowbase;
        if (c + 1 < C_)
            __builtin_prefetch(xrow + HW_ + wq * 4, 0, 0);  // next channel's row
        const float4 xv  = *(const float4*)(xrow + wq * 4);
        const float  ah  = a[(b * C_ + c) * L_ + h];
        const float4 awv = *(const float4*)(a + (b * C_ + c) * L_ + 256 + wq * 4);
        float4 o;
        o.x = xv.x * ah * awv.x * sa4[0];
        o.y = xv.y * ah * awv.y * sa4[1];
        o.z = xv.z * ah * awv.z * sa4[2];
        o.w = xv.w * ah * awv.w * sa4[3];
        *(float4*)(out + rowbase + wq * 4) = o;
    }
}

// ---------------------------------------------------------------------------
extern "C" void kernel_launch(void* const* d_in, const int* in_sizes, int n_in,
                              void* d_out, int out_size, void* d_ws, size_t ws_size,
                              hipStream_t stream) {
    const float* x     = (const float*)d_in[0];
    const float* w01   = (const float*)d_in[1];
    const float* w02   = (const float*)d_in[2];
    const float* gamma = (const float*)d_in[3];
    const float* beta  = (const float*)d_in[4];
    const float* bmean = (const float*)d_in[5];
    const float* bvar  = (const float*)d_in[6];
    const float* wsa   = (const float*)d_in[7];
    float* out = (float*)d_out;

    float* ws     = (float*)d_ws;
    float* rowsum = ws;                       // 262144
    float* rowmax = rowsum + B_*C_*H_;        // 262144
    float* colsum = rowmax + B_*C_*H_;        // 262144
    float* colmax = colsum + B_*C_*W_;        // 262144
    float* pool   = colmax + B_*C_*W_;        // 4194304
    float* y2     = pool   + B_*2*HW_;        // 524288
    float* a      = y2     + B_*C_*L_;        // 524288

    init_kernel      <<<1024, 256, 0, stream>>>(colsum, colmax);
    reduce_kernel    <<<B_*16, 256, 0, stream>>>(x, rowsum, rowmax, colsum, colmax, pool);
    conv01_wmma_kernel<<<B_*8, 256, 0, stream>>>(w01, rowsum, rowmax, colsum, colmax, y2);
    post_kernel      <<<B_*C_, 256, 0, stream>>>(y2, w02, gamma, beta, bmean, bvar, a);
    fuse_out_kernel  <<<B_*(H_/4), 256, 0, stream>>>(x, a, pool, wsa, out);
}